// X_BERT_VQVAE_16999480558083
// MI455X (gfx1250) — compile-verified
//
#include <hip/hip_runtime.h>

typedef float v2f __attribute__((ext_vector_type(2)));
typedef float v8f __attribute__((ext_vector_type(8)));

#define N_ROWS         100352        // 128 * 784
#define DDIM           256
#define KCODES         1024
#define ROWS_PER_BLOCK 128
#define NBLOCKS        (N_ROWS / ROWS_PER_BLOCK)   // 784
#define SB_STRIDE      260           // 256 + 4 pad dwords -> conflict-free b64 reads

// ---------------------------------------------------------------------------
// Kernel 0: per-code squared norms ||c_k||^2
// ---------------------------------------------------------------------------
__global__ void vq_cnorm_kernel(const float* __restrict__ cb,
                                float* __restrict__ cnorm) {
    int k = blockIdx.x * blockDim.x + threadIdx.x;
    if (k >= KCODES) return;
    const float4* row = (const float4*)(cb + (size_t)k * DDIM);
    float s = 0.f;
#pragma unroll 8
    for (int i = 0; i < DDIM / 4; ++i) {
        float4 v = row[i];
        s += v.x * v.x + v.y * v.y + v.z * v.z + v.w * v.w;
    }
    cnorm[k] = s;
}

// ---------------------------------------------------------------------------
// Kernel 1: fused f32-WMMA GEMM + argmin.
// Block = 256 threads = 8 waves; wave w owns rows [blk*128 + w*16, +16).
// A-tile (16x256 f32) lives in registers as 64 fragments of 16x4.
// Codebook streamed 16 codes at a time through padded LDS.
// dist ~ ||c||^2 - 2 z.c  (||z||^2 constant per row, irrelevant for argmin)
// ---------------------------------------------------------------------------
__global__ __launch_bounds__(256)
void vq_argmin_kernel(const float* __restrict__ z,
                      const float* __restrict__ cb,
                      const float* __restrict__ cnorm,
                      int* __restrict__ idx_out,
                      float* __restrict__ idxf_out) {
    __shared__ float sB[16 * SB_STRIDE];
    __shared__ float sCn[KCODES];

    const int tid  = threadIdx.x;
    const int lane = tid & 31;
    const int wave = tid >> 5;
    const int half = lane >> 4;      // 0: lanes 0-15, 1: lanes 16-31
    const int l15  = lane & 15;
    const int koff = half * 2;       // K sub-offset within a 16x4 fragment

    for (int i = tid; i < KCODES; i += 256) sCn[i] = cnorm[i];

    const int row_base = blockIdx.x * ROWS_PER_BLOCK + wave * 16;

    // Load A fragments: lane layout per ISA (lanes 0-15: K=0,1; 16-31: K=2,3)
    v2f afrag[64];
    {
        const float* zr = z + (size_t)(row_base + l15) * DDIM + koff;
#pragma unroll
        for (int kk = 0; kk < 64; ++kk)
            afrag[kk] = *(const v2f*)(zr + kk * 4);
    }

    float best[8];
    int   bidx[8];
#pragma unroll
    for (int j = 0; j < 8; ++j) { best[j] = 3.4e38f; bidx[j] = 0; }

    for (int k0 = 0; k0 < KCODES; k0 += 16) {
        __syncthreads();   // previous tile fully consumed
        {
            // Stage 16x256 codebook tile, padded rows: thread t -> row t/16,
            // 16 contiguous floats starting at (t%16)*16 (4x float4).
            const int r  = tid >> 4;
            const int c0 = (tid & 15) * 16;
            const float4* src = (const float4*)(cb + (size_t)(k0 + r) * DDIM + c0);
            float* dst = &sB[r * SB_STRIDE + c0];
#pragma unroll
            for (int q = 0; q < 4; ++q)
                *(float4*)(dst + q * 4) = src[q];
        }
        __syncthreads();

        v8f acc = {0.f, 0.f, 0.f, 0.f, 0.f, 0.f, 0.f, 0.f};
        const float* brow = &sB[l15 * SB_STRIDE + koff];  // l15 = code column n
#pragma unroll
        for (int kk = 0; kk < 64; ++kk) {
            v2f bfrag = *(const v2f*)(brow + kk * 4);      // ds_load_b64
            acc = __builtin_amdgcn_wmma_f32_16x16x4_f32(
                false, afrag[kk], false, bfrag, (short)0, acc, false, false);
        }

        // C/D layout: VGPR j: lanes 0-15 -> row j, lanes 16-31 -> row j+8;
        // lane's column n = l15 in both halves.
        const float cn   = sCn[k0 + l15];
        const int   kidx = k0 + l15;
#pragma unroll
        for (int j = 0; j < 8; ++j) {
            float v = cn - 2.0f * acc[j];
            if (v < best[j]) { best[j] = v; bidx[j] = kidx; }  // strict < keeps first
        }
    }

    // Cross-lane argmin within each 16-lane half (xor 1,2,4,8 stay in-half).
#pragma unroll
    for (int j = 0; j < 8; ++j) {
        float v = best[j];
        int   i = bidx[j];
#pragma unroll
        for (int off = 8; off >= 1; off >>= 1) {
            float ov = __shfl_xor(v, off, 32);
            int   oi = __shfl_xor(i, off, 32);
            if (ov < v || (ov == v && oi < i)) { v = ov; i = oi; }  // first-occurrence
        }
        if (l15 == 0) {
            int r = row_base + j + half * 8;
            idx_out[r]  = i;
            idxf_out[r] = (float)i;
        }
    }
}

// ---------------------------------------------------------------------------
// Kernel 2: gather z_q = codebook[idx], write straight-through output,
// and one deterministic partial sum of (z - z_q)^2 per block.
// ---------------------------------------------------------------------------
__global__ __launch_bounds__(256)
void vq_gather_kernel(const float* __restrict__ z,
                      const float* __restrict__ cb,
                      const int* __restrict__ idx,
                      float* __restrict__ zq,
                      float* __restrict__ partials) {
    __shared__ float wsum[8];
    const int tid  = threadIdx.x;
    const int lane = tid & 31;
    const int wave = tid >> 5;
    const int row0 = blockIdx.x * ROWS_PER_BLOCK + wave * 16;

    float acc = 0.f;
    for (int rr = 0; rr < 16; ++rr) {
        const int row = row0 + rr;
        const int k   = idx[row];
        const float4* cr = (const float4*)(cb + (size_t)k   * DDIM);
        const float4* zr = (const float4*)(z  + (size_t)row * DDIM);
        float4*       qr = (float4*)      (zq + (size_t)row * DDIM);
        float4 c0 = cr[lane], c1 = cr[lane + 32];
        float4 z0 = zr[lane], z1 = zr[lane + 32];
        qr[lane]      = c0;
        qr[lane + 32] = c1;
        float d;
        d = z0.x - c0.x; acc += d * d;
        d = z0.y - c0.y; acc += d * d;
        d = z0.z - c0.z; acc += d * d;
        d = z0.w - c0.w; acc += d * d;
        d = z1.x - c1.x; acc += d * d;
        d = z1.y - c1.y; acc += d * d;
        d = z1.z - c1.z; acc += d * d;
        d = z1.w - c1.w; acc += d * d;
    }
#pragma unroll
    for (int off = 16; off >= 1; off >>= 1)
        acc += __shfl_xor(acc, off, 32);
    if (lane == 0) wsum[wave] = acc;
    __syncthreads();
    if (tid == 0) {
        float s = 0.f;
#pragma unroll
        for (int w = 0; w < 8; ++w) s += wsum[w];
        partials[blockIdx.x] = s;
    }
}

// ---------------------------------------------------------------------------
// Kernel 3: reduce 784 partials; loss_vq == loss_commit == mean((z - z_q)^2)
// ---------------------------------------------------------------------------
__global__ void vq_finalize_kernel(const float* __restrict__ partials,
                                   float* __restrict__ out01) {
    __shared__ float red[256];
    float s = 0.f;
    for (int i = threadIdx.x; i < NBLOCKS; i += 256) s += partials[i];
    red[threadIdx.x] = s;
    __syncthreads();
    for (int off = 128; off >= 1; off >>= 1) {
        if (threadIdx.x < off) red[threadIdx.x] += red[threadIdx.x + off];
        __syncthreads();
    }
    if (threadIdx.x == 0) {
        float mean = red[0] / (float)((size_t)N_ROWS * DDIM);
        out01[0] = mean;   // loss_vq
        out01[1] = mean;   // loss_commit (identical value)
    }
}

// ---------------------------------------------------------------------------
extern "C" void kernel_launch(void* const* d_in, const int* in_sizes, int n_in,
                              void* d_out, int out_size, void* d_ws, size_t ws_size,
                              hipStream_t stream) {
    const float* z  = (const float*)d_in[0];   // [128,784,256]
    const float* cb = (const float*)d_in[1];   // [1024,256]

    float* out    = (float*)d_out;
    float* loss01 = out;                                   // [0]=loss_vq [1]=loss_commit
    float* zq     = out + 2;                               // [N_ROWS * DDIM]
    float* idxf   = out + 2 + (size_t)N_ROWS * DDIM;       // [N_ROWS] indices (as float)

    char*  ws       = (char*)d_ws;
    float* cnorm    = (float*)ws;                                   // 1024 f32
    int*   idx      = (int*)(ws + 4096);                            // N_ROWS i32
    float* partials = (float*)(ws + 4096 + (size_t)N_ROWS * 4);     // NBLOCKS f32

    vq_cnorm_kernel<<<(KCODES + 255) / 256, 256, 0, stream>>>(cb, cnorm);
    vq_argmin_kernel<<<NBLOCKS, 256, 0, stream>>>(z, cb, cnorm, idx, idxf);
    vq_gather_kernel<<<NBLOCKS, 256, 0, stream>>>(z, cb, idx, zq, partials);
    vq_finalize_kernel<<<1, 256, 0, stream>>>(partials, loss01);
}